// SSALayer_78769700208930
// MI455X (gfx1250) — compile-verified
//
#include <hip/hip_runtime.h>
#include <math.h>

#define NN 1024
#define MM 1024
#define DD 256

typedef __attribute__((ext_vector_type(2))) float v2f;
typedef __attribute__((ext_vector_type(8))) float v8f;

// ---------------------------------------------------------------------------
// K1: S[i,j] = -sum_d |zx[i,d] - zy[j,d]| ; 64x64 tile per 256-thread block,
// 4x4 register blocking, d-chunks of 32 staged in LDS (stride 33 = conflict
// free for the strided ys reads).
// ---------------------------------------------------------------------------
__global__ __launch_bounds__(256) void k1_dist(const float* __restrict__ zx,
                                               const float* __restrict__ zy,
                                               float* __restrict__ S) {
  __shared__ float xs[64][33];
  __shared__ float ys[64][33];
  const int tid = threadIdx.x;
  const int tx = tid & 15, ty = tid >> 4;
  const int i0 = blockIdx.y * 64, j0 = blockIdx.x * 64;

  float acc[4][4];
#pragma unroll
  for (int a = 0; a < 4; ++a)
#pragma unroll
    for (int b = 0; b < 4; ++b) acc[a][b] = 0.f;

  for (int d0 = 0; d0 < DD; d0 += 32) {
    __syncthreads();
#pragma unroll
    for (int k = 0; k < 8; ++k) {
      int e = tid + k * 256;
      int r = e >> 5, c = e & 31;
      xs[r][c] = zx[(size_t)(i0 + r) * DD + d0 + c];
      ys[r][c] = zy[(size_t)(j0 + r) * DD + d0 + c];
    }
    __syncthreads();
#pragma unroll
    for (int dd = 0; dd < 32; ++dd) {
      float xv[4], yv[4];
#pragma unroll
      for (int a = 0; a < 4; ++a) xv[a] = xs[ty + 16 * a][dd];
#pragma unroll
      for (int b = 0; b < 4; ++b) yv[b] = ys[tx + 16 * b][dd];
#pragma unroll
      for (int a = 0; a < 4; ++a)
#pragma unroll
        for (int b = 0; b < 4; ++b)
          acc[a][b] += __builtin_fabsf(xv[a] - yv[b]);
    }
  }

#pragma unroll
  for (int a = 0; a < 4; ++a)
#pragma unroll
    for (int b = 0; b < 4; ++b)
      S[(size_t)(i0 + ty + 16 * a) * MM + (j0 + tx + 16 * b)] = -acc[a][b];
}

// ---------------------------------------------------------------------------
// K2a: per-row max and exp-sum. One block per row.
// ---------------------------------------------------------------------------
__global__ __launch_bounds__(256) void k2_rows(const float* __restrict__ S,
                                               float* __restrict__ rowmax,
                                               float* __restrict__ rowsum) {
  __shared__ float red[256];
  const int i = blockIdx.x;
  const int tid = threadIdx.x;
  float v[4];
#pragma unroll
  for (int k = 0; k < 4; ++k) v[k] = S[(size_t)i * MM + tid + 256 * k];
  float m = fmaxf(fmaxf(v[0], v[1]), fmaxf(v[2], v[3]));
  red[tid] = m;
  __syncthreads();
  for (int s = 128; s > 0; s >>= 1) {
    if (tid < s) red[tid] = fmaxf(red[tid], red[tid + s]);
    __syncthreads();
  }
  m = red[0];
  __syncthreads();
  float p = 0.f;
#pragma unroll
  for (int k = 0; k < 4; ++k) p += __expf(v[k] - m);
  red[tid] = p;
  __syncthreads();
  for (int s = 128; s > 0; s >>= 1) {
    if (tid < s) red[tid] += red[tid + s];
    __syncthreads();
  }
  if (tid == 0) {
    rowmax[i] = m;
    rowsum[i] = red[0];
  }
}

// ---------------------------------------------------------------------------
// K2b: per-column max + exp-sum over rows for a 16-column tile, one wave per
// block. Column sums computed exactly in f32 by chaining
// V_WMMA_F32_16X16X4_F32 with A = ones(16x4):  C += ones x E_chunk(4x16).
// B layout per ISA: lanes 0-15 hold K=0 (VGPR0) / K=1 (VGPR1) for N=lane,
// lanes 16-31 hold K=2 / K=3.
// ---------------------------------------------------------------------------
__global__ __launch_bounds__(32) void k2_cols(const float* __restrict__ S,
                                              float* __restrict__ colmax,
                                              float* __restrict__ colsum) {
  const int lane = threadIdx.x;
  const int half = lane >> 4;      // 0: K rows 0,1   1: K rows 2,3
  const int jl = lane & 15;
  const int j = blockIdx.x * 16 + jl;

  // pass 1: column max (each lane sees 2 of every 4 rows; halves combined)
  float m = -3.402823466e38f;
  for (int base = 0; base < NN; base += 4) {
    int r = base + half * 2;
    float s0 = S[(size_t)r * MM + j];
    float s1 = S[(size_t)(r + 1) * MM + j];
    m = fmaxf(m, fmaxf(s0, s1));
  }
  m = fmaxf(m, __shfl_xor(m, 16, 32));  // combine halves -> full column max

  // pass 2: chained WMMA accumulation of exp(s - colmax)
  v2f afrag;
  afrag.x = 1.0f;
  afrag.y = 1.0f;
  v8f cacc = {};
  for (int base = 0; base < NN; base += 4) {
    int r = base + half * 2;
    float s0 = S[(size_t)r * MM + j];
    float s1 = S[(size_t)(r + 1) * MM + j];
    v2f bfrag;
    bfrag.x = __expf(s0 - m);
    bfrag.y = __expf(s1 - m);
    cacc = __builtin_amdgcn_wmma_f32_16x16x4_f32(
        /*neg_a=*/false, afrag, /*neg_b=*/false, bfrag,
        /*c_mod=*/(short)0, cacc, /*reuse_a=*/false, /*reuse_b=*/false);
  }

  if (lane < 16) {
    colmax[j] = m;
    colsum[j] = cacc[0];  // C row 0, N = lane
  }
}

// ---------------------------------------------------------------------------
// K3: per-row partial sums of w*s and w, w = a + b - a*b.
// ---------------------------------------------------------------------------
__global__ __launch_bounds__(256) void k3_weight(
    const float* __restrict__ S, const float* __restrict__ rowmax,
    const float* __restrict__ rowsum, const float* __restrict__ colmax,
    const float* __restrict__ colsum, float* __restrict__ partials) {
  __shared__ float rn[256];
  __shared__ float rd[256];
  const int i = blockIdx.x;
  const int tid = threadIdx.x;
  const float rm = rowmax[i];
  const float rinv = 1.0f / rowsum[i];
  float num = 0.f, den = 0.f;
#pragma unroll
  for (int k = 0; k < 4; ++k) {
    int jj = tid + 256 * k;
    float s = S[(size_t)i * MM + jj];
    float a = __expf(s - rm) * rinv;
    float b = __expf(s - colmax[jj]) / colsum[jj];
    float w = a + b - a * b;
    num += w * s;
    den += w;
  }
  rn[tid] = num;
  rd[tid] = den;
  __syncthreads();
  for (int s2 = 128; s2 > 0; s2 >>= 1) {
    if (tid < s2) {
      rn[tid] += rn[tid + s2];
      rd[tid] += rd[tid + s2];
    }
    __syncthreads();
  }
  if (tid == 0) {
    partials[2 * i] = rn[0];
    partials[2 * i + 1] = rd[0];
  }
}

// ---------------------------------------------------------------------------
// K4: reduce 1024 (num,den) pairs, write scalar c = num/den.
// ---------------------------------------------------------------------------
__global__ __launch_bounds__(256) void k4_final(const float* __restrict__ partials,
                                                float* __restrict__ out) {
  __shared__ float rn[256];
  __shared__ float rd[256];
  const int tid = threadIdx.x;
  float num = 0.f, den = 0.f;
#pragma unroll
  for (int k = 0; k < 4; ++k) {
    num += partials[2 * (tid + 256 * k)];
    den += partials[2 * (tid + 256 * k) + 1];
  }
  rn[tid] = num;
  rd[tid] = den;
  __syncthreads();
  for (int s = 128; s > 0; s >>= 1) {
    if (tid < s) {
      rn[tid] += rn[tid + s];
      rd[tid] += rd[tid + s];
    }
    __syncthreads();
  }
  if (tid == 0) out[0] = rn[0] / rd[0];
}

extern "C" void kernel_launch(void* const* d_in, const int* in_sizes, int n_in,
                              void* d_out, int out_size, void* d_ws,
                              size_t ws_size, hipStream_t stream) {
  const float* zx = (const float*)d_in[0];
  const float* zy = (const float*)d_in[1];
  float* out = (float*)d_out;

  float* ws = (float*)d_ws;
  float* S = ws;                               // 1024*1024
  float* rowmax = S + (size_t)NN * MM;         // 1024
  float* rowsum = rowmax + NN;                 // 1024
  float* colmax = rowsum + NN;                 // 1024
  float* colsum = colmax + MM;                 // 1024
  float* partials = colsum + MM;               // 2048

  dim3 g1(MM / 64, NN / 64);
  k1_dist<<<g1, 256, 0, stream>>>(zx, zy, S);
  k2_rows<<<NN, 256, 0, stream>>>(S, rowmax, rowsum);
  k2_cols<<<MM / 16, 32, 0, stream>>>(S, colmax, colsum);
  k3_weight<<<NN, 256, 0, stream>>>(S, rowmax, rowsum, colmax, colsum, partials);
  k4_final<<<1, 256, 0, stream>>>(partials, out);
}